// GCNEncoder_48979807044073
// MI455X (gfx1250) — compile-verified
//
#include <hip/hip_runtime.h>
#include <hip/hip_bf16.h>
#include <math.h>

typedef __attribute__((ext_vector_type(2))) float v2f;
typedef __attribute__((ext_vector_type(8))) float v8f;

#define N_NODES 50000
#define N_EDGES 800000
#define IN_DIM  128
#define HID     64
#define ZDIM    64
#define N_GRAPHS 64

// ---------------------------------------------------------------- utilities
__global__ void fill_kernel(float* __restrict__ p, long n, float v) {
    long i = (long)blockIdx.x * blockDim.x + threadIdx.x;
    long stride = (long)gridDim.x * blockDim.x;
    for (; i < n; i += stride) p[i] = v;
}

__global__ void deg_kernel(const int* __restrict__ dst, float* __restrict__ deg, int E) {
    int e = blockIdx.x * blockDim.x + threadIdx.x;
    if (e < E) atomicAdd(&deg[dst[e]], 1.0f);
}

__global__ void rsqrt_kernel(float* __restrict__ d, int n) {
    int i = blockIdx.x * blockDim.x + threadIdx.x;
    if (i < n) d[i] = rsqrtf(d[i]);   // deg >= 1 always (self loop)
}

// ------------------------------------------------- GEMM1: g1 = (x @ W1) * dinv
// grid = 3125 blocks (16 rows each), 128 threads = 4 waves, wave w -> cols [16w,16w+16)
__global__ void gemm1_wmma(const float* __restrict__ x, const float* __restrict__ W,
                           const float* __restrict__ dinv, float* __restrict__ g1) {
    const int wave  = threadIdx.x >> 5;
    const int lane  = threadIdx.x & 31;
    const int lhalf = lane >> 4;     // 0: lanes 0-15, 1: lanes 16-31
    const int l16   = lane & 15;
    const int row0  = blockIdx.x * 16;
    const int col0  = wave * 16;

    v8f acc = {};
    const float* xrow = x + (long)(row0 + l16) * IN_DIM;
    #pragma unroll 4
    for (int k = 0; k < IN_DIM; k += 4) {
        // A 16x4 f32: lane l16 = row; VGPR0/1 = {k+2*lhalf, k+2*lhalf+1}
        v2f a, b;
        const float2 av = *(const float2*)(xrow + k + 2 * lhalf);
        a[0] = av.x; a[1] = av.y;
        // B 4x16 f32: lane l16 = col; VGPR0/1 = rows {k+2*lhalf, k+2*lhalf+1}
        const float* wp = W + (long)(k + 2 * lhalf) * HID + col0 + l16;
        b[0] = wp[0]; b[1] = wp[HID];
        acc = __builtin_amdgcn_wmma_f32_16x16x4_f32(false, a, false, b,
                                                    (short)0, acc, false, false);
    }
    // D: VGPR r, lane<16 -> (row0+r, col0+lane); lane>=16 -> (row0+r+8, col0+l16)
    #pragma unroll
    for (int r = 0; r < 8; ++r) {
        int row = row0 + r + lhalf * 8;
        g1[(long)row * HID + col0 + l16] = acc[r] * dinv[row];
    }
}

// -------------------- GEMM2 (fused mu|sigma): g2[:,0:64]=(h@Wmu)*dinv, [:,64:128]=(h@Wsig)*dinv
// grid = 3125 blocks, 256 threads = 8 waves: waves 0-3 -> Wmu, 4-7 -> Wsig
__global__ void gemm2_wmma(const float* __restrict__ h, const float* __restrict__ Wmu,
                           const float* __restrict__ Wsig, const float* __restrict__ dinv,
                           float* __restrict__ g2) {
    const int wave  = threadIdx.x >> 5;
    const int lane  = threadIdx.x & 31;
    const int lhalf = lane >> 4;
    const int l16   = lane & 15;
    const int row0  = blockIdx.x * 16;
    const float* W  = (wave < 4) ? Wmu : Wsig;
    const int colW  = (wave & 3) * 16;          // column inside the 64-wide weight
    const int colO  = (wave < 4 ? 0 : ZDIM) + colW; // column inside 128-wide output

    v8f acc = {};
    const float* hrow = h + (long)(row0 + l16) * HID;
    #pragma unroll 4
    for (int k = 0; k < HID; k += 4) {
        v2f a, b;
        const float2 av = *(const float2*)(hrow + k + 2 * lhalf);
        a[0] = av.x; a[1] = av.y;
        const float* wp = W + (long)(k + 2 * lhalf) * ZDIM + colW + l16;
        b[0] = wp[0]; b[1] = wp[ZDIM];
        acc = __builtin_amdgcn_wmma_f32_16x16x4_f32(false, a, false, b,
                                                    (short)0, acc, false, false);
    }
    #pragma unroll
    for (int r = 0; r < 8; ++r) {
        int row = row0 + r + lhalf * 8;
        g2[(long)row * (2 * ZDIM) + colO + l16] = acc[r] * dinv[row];
    }
}

// ----------------------------------------------- scatter: acc[d] += g[s] (unscaled)
// one wave per edge, 64 features, 2 per lane
__global__ void scatter64(const int* __restrict__ src, const int* __restrict__ dst,
                          const float* __restrict__ g, float* __restrict__ acc, int E) {
    int wid  = (blockIdx.x * blockDim.x + threadIdx.x) >> 5;
    int lane = threadIdx.x & 31;
    if (wid >= E) return;
    int s = src[wid], d = dst[wid];
    const float2 v = *(const float2*)(g + (long)s * HID + lane * 2);
    float* out = acc + (long)d * HID + lane * 2;
    atomicAdd(out,     v.x);
    atomicAdd(out + 1, v.y);
}

// one wave per edge, 128 features, 4 per lane
__global__ void scatter128(const int* __restrict__ src, const int* __restrict__ dst,
                           const float* __restrict__ g, float* __restrict__ acc, int E) {
    int wid  = (blockIdx.x * blockDim.x + threadIdx.x) >> 5;
    int lane = threadIdx.x & 31;
    if (wid >= E) return;
    int s = src[wid], d = dst[wid];
    const float4 v = *(const float4*)(g + (long)s * 128 + lane * 4);
    float* out = acc + (long)d * 128 + lane * 4;
    atomicAdd(out,     v.x);
    atomicAdd(out + 1, v.y);
    atomicAdd(out + 2, v.z);
    atomicAdd(out + 3, v.w);
}

// ---------------------------------- epilogue 1: h = ELU(dinv*(acc+g) + b1), in-place OK
__global__ void epi1_kernel(const float* __restrict__ acc, const float* __restrict__ g,
                            const float* __restrict__ dinv, const float* __restrict__ b1,
                            float* __restrict__ h) {
    long idx = (long)blockIdx.x * blockDim.x + threadIdx.x;
    if (idx >= (long)N_NODES * HID) return;
    int i = (int)(idx >> 6), f = (int)(idx & 63);
    float v = dinv[i] * (acc[idx] + g[idx]) + b1[f];
    h[idx] = v > 0.0f ? v : expm1f(v);
}

// ---------------- epilogue 2: ELU + segment-sum pooling into zsum[64][128], zcnt[64]
__global__ void epi2_pool_kernel(const float* __restrict__ acc, const float* __restrict__ g,
                                 const float* __restrict__ dinv, const float* __restrict__ bmu,
                                 const float* __restrict__ bsig, const int* __restrict__ batch,
                                 float* __restrict__ zsum, float* __restrict__ zcnt) {
    long idx = (long)blockIdx.x * blockDim.x + threadIdx.x;
    if (idx >= (long)N_NODES * 128) return;
    int i = (int)(idx >> 7), f = (int)(idx & 127);
    float bias = (f < ZDIM) ? bmu[f] : bsig[f - ZDIM];
    float v = dinv[i] * (acc[idx] + g[idx]) + bias;
    v = v > 0.0f ? v : expm1f(v);
    int gr = batch[i];
    atomicAdd(&zsum[gr * 128 + f], v);
    if (f == 0) atomicAdd(&zcnt[gr], 1.0f);
}

// ------------------------------------- finalize: mean + split into (z_mu, z_sigma)
__global__ void finalize_kernel(const float* __restrict__ zsum, const float* __restrict__ zcnt,
                                float* __restrict__ out) {
    int idx = blockIdx.x * blockDim.x + threadIdx.x;
    if (idx >= N_GRAPHS * 128) return;
    int gr = idx >> 7, f = idx & 127;
    float c = fmaxf(zcnt[gr], 1.0f);
    float v = zsum[idx] / c;
    if (f < ZDIM) out[gr * ZDIM + f] = v;
    else          out[N_GRAPHS * ZDIM + gr * ZDIM + (f - ZDIM)] = v;
}

extern "C" void kernel_launch(void* const* d_in, const int* in_sizes, int n_in,
                              void* d_out, int out_size, void* d_ws, size_t ws_size,
                              hipStream_t stream) {
    const float* x     = (const float*)d_in[0];
    const int*   ei    = (const int*)d_in[1];          // [2, E] row-major
    const int*   batch = (const int*)d_in[2];
    const float* W1    = (const float*)d_in[4];
    const float* b1    = (const float*)d_in[5];
    const float* Wmu   = (const float*)d_in[6];
    const float* bmu   = (const float*)d_in[7];
    const float* Wsig  = (const float*)d_in[8];
    const float* bsig  = (const float*)d_in[9];
    float* out = (float*)d_out;

    const int E = in_sizes[1] / 2;
    const int* src = ei;
    const int* dst = ei + E;

    // workspace layout (floats)
    float* ws   = (float*)d_ws;
    long off = 0;
    float* dinv = ws + off; off += 50048;                       // deg -> dinv in place
    float* g1   = ws + off; off += (long)N_NODES * HID;         // pre-scaled x@W1
    float* acc1 = ws + off; off += (long)N_NODES * HID;         // scatter acc, then h
    float* g2   = ws + off; off += (long)N_NODES * 128;         // pre-scaled [h@Wmu | h@Wsig]
    float* acc2 = ws + off; off += (long)N_NODES * 128;
    float* zsum = ws + off; off += N_GRAPHS * 128;
    float* zcnt = ws + off; off += N_GRAPHS;
    (void)ws_size; (void)n_in; (void)out_size;

    const int T = 256;
    // init (must re-zero every call: graph replay reuses buffers)
    fill_kernel<<<1024, T, 0, stream>>>(dinv, N_NODES, 1.0f);                  // self-loop deg
    fill_kernel<<<4096, T, 0, stream>>>(acc1, (long)N_NODES * HID, 0.0f);
    fill_kernel<<<4096, T, 0, stream>>>(acc2, (long)N_NODES * 128, 0.0f);
    fill_kernel<<<64,   T, 0, stream>>>(zsum, N_GRAPHS * 128 + N_GRAPHS, 0.0f); // zsum+zcnt contiguous

    // degrees -> dinv
    deg_kernel<<<(E + T - 1) / T, T, 0, stream>>>(dst, dinv, E);
    rsqrt_kernel<<<(N_NODES + T - 1) / T, T, 0, stream>>>(dinv, N_NODES);

    // layer 1
    gemm1_wmma<<<N_NODES / 16, 128, 0, stream>>>(x, W1, dinv, g1);
    scatter64<<<(E * 32 + T - 1) / T, T, 0, stream>>>(src, dst, g1, acc1, E);
    epi1_kernel<<<((long)N_NODES * HID + T - 1) / T, T, 0, stream>>>(acc1, g1, dinv, b1, acc1);

    // layers 2+3 fused (h lives in acc1)
    gemm2_wmma<<<N_NODES / 16, 256, 0, stream>>>(acc1, Wmu, Wsig, dinv, g2);
    scatter128<<<(E * 32 + T - 1) / T, T, 0, stream>>>(src, dst, g2, acc2, E);
    epi2_pool_kernel<<<((long)N_NODES * 128 + T - 1) / T, T, 0, stream>>>(
        acc2, g2, dinv, bmu, bsig, batch, zsum, zcnt);

    finalize_kernel<<<(N_GRAPHS * 128 + T - 1) / T, T, 0, stream>>>(zsum, zcnt, out);
}